// DiffusionModel_11501922418758
// MI455X (gfx1250) — compile-verified
//
#include <hip/hip_runtime.h>
#include <math.h>

typedef __attribute__((ext_vector_type(2))) float v2f;
typedef __attribute__((ext_vector_type(8))) float v8f;

#define BATCH 16
#define NPTS  2048
#define NDIM  3
#define NT    1024          // 32 waves (wave32)
#define WAVES (NT / 32)     // 32
#define ROWTILES (NPTS / 16) // 128
#define RT_PER_WAVE (ROWTILES / WAVES) // 4
#define ITERS 5
#define EPSB  0.001f
#define NEG_INF_BITS ((int)0xFF800000)

__global__ __launch_bounds__(NT) void auction_flow_kernel(
    const float* __restrict__ cloud,
    const float* __restrict__ noise,
    const float* __restrict__ tin,
    float* __restrict__ out)
{
    // B-fragment halves: lanes 0-15 read blo = (x, y) [K=0,1],
    //                    lanes 16-31 read bhi = (z, yn + price) [K=2,3]
    __shared__ float2 blo[NPTS];
    __shared__ float2 bhi[NPTS];
    __shared__ int    nprice[NPTS];  // fp32 bid bits, int atomicMax (bids > 0)
    __shared__ int    bestIdx[NPTS];
    __shared__ float  s_inv;

    const int b    = blockIdx.x;
    const int tid  = threadIdx.x;
    const int lane = tid & 31;
    const int wave = tid >> 5;
    const int half = lane >> 4;
    const int l15  = lane & 15;

    const float* cb = cloud + (size_t)b * NPTS * NDIM;
    const float* nb = noise + (size_t)b * NPTS * NDIM;

    // ---------- per-batch std (ddof=1) over N*D elements ----------
    float s = 0.f, s2 = 0.f;
    for (int k = tid; k < NPTS * NDIM; k += NT) {
        float x = cb[k];
        s += x; s2 += x * x;
    }
    float* r1 = (float*)nprice;      // reuse LDS as reduce scratch
    float* r2 = (float*)bestIdx;
    r1[tid] = s; r2[tid] = s2;
    __syncthreads();
    for (int off = NT / 2; off > 0; off >>= 1) {
        if (tid < off) { r1[tid] += r1[tid + off]; r2[tid] += r2[tid + off]; }
        __syncthreads();
    }
    if (tid == 0) {
        float sum = r1[0], sumsq = r2[0];
        const float n = (float)(NPTS * NDIM);
        float var = (sumsq - sum * sum / n) / (n - 1.0f);
        s_inv = rsqrtf(var);
    }
    __syncthreads();
    const float inv_std = s_inv;

    // ---------- build x0 fragments in LDS (price starts at 0) ----------
    for (int i = tid; i < NPTS; i += NT) {
        float x = cb[i * 3 + 0] * inv_std;
        float y = cb[i * 3 + 1] * inv_std;
        float z = cb[i * 3 + 2] * inv_std;
        blo[i] = make_float2(x, y);
        bhi[i] = make_float2(z, x * x + y * y + z * z);  // yn + price(=0)
    }
    __syncthreads();

    // ---------- precompute the wave's 4 A fragments (registers) ----------
    // A row i (K=0..3) = (-2nx, -2ny, -2nz, 1):  C = (yn_j + price_j) - 2*cross_ij
    v2f afrag[RT_PER_WAVE];
#pragma unroll
    for (int k = 0; k < RT_PER_WAVE; ++k) {
        int m = (wave + k * WAVES) * 16 + l15;
        float nx = nb[m * 3 + 0];
        float ny = nb[m * 3 + 1];
        float nz = nb[m * 3 + 2];
        afrag[k].x = (half == 0) ? (-2.f * nx) : (-2.f * nz);
        afrag[k].y = (half == 0) ? (-2.f * ny) : 1.0f;
    }

    // per-lane B base pointer (selected once; divergent address is fine)
    const v2f* bsel = (half == 0) ? (const v2f*)blo : (const v2f*)bhi;
    const float FINF = __int_as_float(0x7F800000);

    for (int iter = 0; iter < ITERS; ++iter) {
        for (int i = tid; i < NPTS; i += NT) nprice[i] = NEG_INF_BITS;
        __syncthreads();

#pragma unroll
        for (int k = 0; k < RT_PER_WAVE; ++k) {
            const int rt = wave + k * WAVES;
            const v2f a = afrag[k];

            float m1[8], m2[8]; int i1[8];
#pragma unroll
            for (int r = 0; r < 8; ++r) { m1[r] = FINF; m2[r] = FINF; i1[r] = 0; }

#pragma unroll 4
            for (int jt = 0; jt < NPTS / 16; ++jt) {
                const int j = jt * 16 + l15;
                v2f bf = bsel[j];
                v8f c = {};
                c = __builtin_amdgcn_wmma_f32_16x16x4_f32(
                        false, a, false, bf, (short)0, c, false, false);
#pragma unroll
                for (int r = 0; r < 8; ++r) {
                    // branchless top-2 (min) via pure compare+select
                    // (no fminf/fmaxf -> no canonicalize ops)
                    float w   = c[r];
                    bool  lt1 = w < m1[r];
                    bool  lt2 = w < m2[r];
                    float nm2 = lt1 ? m1[r] : (lt2 ? w : m2[r]);
                    float nm1 = lt1 ? w : m1[r];
                    int   ni  = lt1 ? j : i1[r];
                    m1[r] = nm1; m2[r] = nm2; i1[r] = ni;
                }
            }

            // branchless top-2 merge across the 16 lanes of each half
#pragma unroll
            for (int r = 0; r < 8; ++r) {
#pragma unroll
                for (int off = 1; off < 16; off <<= 1) {
                    float o1 = __shfl_xor(m1[r], off, 32);
                    int   oi = __shfl_xor(i1[r], off, 32);
                    float o2 = __shfl_xor(m2[r], off, 32);
                    bool take = (o1 < m1[r]) || (o1 == m1[r] && oi < i1[r]);
                    float a2 = take ? m1[r] : o1;
                    float b2 = take ? o2 : m2[r];
                    float nm2 = (a2 < b2) ? a2 : b2;
                    float nm1 = take ? o1 : m1[r];
                    int   ni  = take ? oi : i1[r];
                    m1[r] = nm1; m2[r] = nm2; i1[r] = ni;
                }
            }
            if (l15 == 0) {
#pragma unroll
                for (int r = 0; r < 8; ++r) {
                    int row = rt * 16 + half * 8 + r;   // C: VGPR r -> rows r / r+8
                    bestIdx[row] = i1[r];
                    float bid = m2[r] - m1[r] + EPSB;   // >= eps > 0
                    atomicMax(&nprice[i1[r]], __float_as_int(bid));
                }
            }
        }
        __syncthreads();
        // price[j] <- max bid if any bidder, else unchanged; folded into bhi.y
        for (int i = tid; i < NPTS; i += NT) {
            int np = nprice[i];
            if (np != NEG_INF_BITS) {
                float2 lo = blo[i];
                float2 hi = bhi[i];
                float yn = lo.x * lo.x + lo.y * lo.y + hi.x * hi.x;
                bhi[i] = make_float2(hi.x, yn + __int_as_float(np));
            }
        }
        __syncthreads();
    }

    // ---------- gather + flow outputs: stack([x_t, v]) ----------
    const float tv = tin[b];
    float* out_xt = out + (size_t)b * NPTS * NDIM;
    float* out_v  = out + (size_t)BATCH * NPTS * NDIM + (size_t)b * NPTS * NDIM;
    for (int i = tid; i < NPTS; i += NT) {
        int idx = bestIdx[i];
        float2 lo = blo[idx];
        float  xz = bhi[idx].x;
        float nx = nb[i * 3 + 0];
        float ny = nb[i * 3 + 1];
        float nz = nb[i * 3 + 2];
        out_xt[i * 3 + 0] = (1.f - tv) * lo.x + tv * nx;
        out_xt[i * 3 + 1] = (1.f - tv) * lo.y + tv * ny;
        out_xt[i * 3 + 2] = (1.f - tv) * xz   + tv * nz;
        out_v[i * 3 + 0]  = nx - lo.x;
        out_v[i * 3 + 1]  = ny - lo.y;
        out_v[i * 3 + 2]  = nz - xz;
    }
}

extern "C" void kernel_launch(void* const* d_in, const int* in_sizes, int n_in,
                              void* d_out, int out_size, void* d_ws, size_t ws_size,
                              hipStream_t stream) {
    (void)in_sizes; (void)n_in; (void)d_ws; (void)ws_size; (void)out_size;
    const float* cloud = (const float*)d_in[0];
    const float* noise = (const float*)d_in[1];
    const float* t     = (const float*)d_in[2];
    float* out = (float*)d_out;
    auction_flow_kernel<<<BATCH, NT, 0, stream>>>(cloud, noise, t, out);
}